// TopKEdgePooling_66357244723900
// MI455X (gfx1250) — compile-verified
//
#include <hip/hip_runtime.h>
#include <hip/hip_bf16.h>
#include <stdint.h>

// ---------------- problem constants ----------------
#define BB      16
#define E_PER   65536          // 2^16
#define E_TOT   (BB * E_PER)   // 1,048,576
#define NN      262144
#define DD      128
#define HID     128
#define KK      32768          // 2^15
#define BK      (BB * KK)      // 524,288
#define LOG_EPER 16
#define LOG_K    15

// LDS tile for the fused bitonic phases: 2048 keys * 8B = 16 KB (of 320 KB/WGP)
#define SORT_TILE 2048

typedef __attribute__((ext_vector_type(2))) float v2f;
typedef __attribute__((ext_vector_type(8))) float v8f;

// float -> sortable uint32, then inverted => ascending uint sort == descending float
__device__ __forceinline__ unsigned flip_desc(float f) {
  unsigned u = __float_as_uint(f);
  unsigned s = (u & 0x80000000u) ? ~u : (u | 0x80000000u);
  return ~s;
}

// ---------------- pass-through copies ----------------
__global__ void copy_f4(const float4* __restrict__ src, float4* __restrict__ dst, int n) {
  int i = blockIdx.x * blockDim.x + threadIdx.x;
  if (i < n) dst[i] = src[i];
}
__global__ void copy_i32(const int* __restrict__ src, int* __restrict__ dst, int n) {
  int i = blockIdx.x * blockDim.x + threadIdx.x;
  if (i < n) dst[i] = src[i];
}
__global__ void zero_i32(int* __restrict__ p, int n) {
  int i = blockIdx.x * blockDim.x + threadIdx.x;
  if (i < n) p[i] = 0;
}

// ---------------- edge MLP via V_WMMA_F32_16X16X4_F32 ----------------
// pi[e] = relu(edge_attr[e]@w1 + b1) @ w2 + b2    (IN_CH=2 padded to K=4)
// One wave handles TPW tiles of 16 edges. B operand (w1 chunk) hoisted in regs.
#define TPW 8
__global__ void edge_mlp_wmma(const float* __restrict__ edge_attr,
                              const float* __restrict__ w1,
                              const float* __restrict__ b1,
                              const float* __restrict__ w2,
                              const float* __restrict__ b2,
                              unsigned long long* __restrict__ keys) {
  const int lane = threadIdx.x & 31;
  const int wave = (blockIdx.x * blockDim.x + threadIdx.x) >> 5;
  const int col  = lane & 15;
  const bool lo  = lane < 16;          // lanes 0-15: K rows 0/1; lanes 16-31: K rows 2/3 (zero pad)

  // B-matrix (4x16 f32, 2 VGPRs): VGPR0 = row K0 (lanes 0-15) / K2 (lanes 16-31); VGPR1 = K1/K3
  v2f   Bv[8];
  float w2n[8], b1n[8];
#pragma unroll
  for (int c = 0; c < 8; ++c) {
    int n = c * 16 + col;
    float r0 = w1[n];            // w1[0][n]
    float r1 = w1[HID + n];      // w1[1][n]
    Bv[c].x = lo ? r0 : 0.0f;
    Bv[c].y = lo ? r1 : 0.0f;
    w2n[c] = w2[n];
    b1n[c] = b1[n];
  }
  const float b2s = b2[0];

  for (int tt = 0; tt < TPW; ++tt) {
    const int e0 = (wave * TPW + tt) * 16;
    // A-matrix (16x4 f32, 2 VGPRs): VGPR0 = K0 (lanes 0-15, M=lane) / K2 (zero); VGPR1 = K1/K3
    const int ea = e0 + col;                 // always in range
    float a0 = edge_attr[2 * ea + 0];
    float a1 = edge_attr[2 * ea + 1];
    v2f A;
    A.x = lo ? a0 : 0.0f;
    A.y = lo ? a1 : 0.0f;

    float pf[8];
#pragma unroll
    for (int i = 0; i < 8; ++i) pf[i] = 0.0f;

#pragma unroll
    for (int c = 0; c < 8; ++c) {
      v8f acc;
#pragma unroll
      for (int i = 0; i < 8; ++i) acc[i] = b1n[c];   // bias broadcast along M
      // 8 args: (neg_a, A, neg_b, B, c_mod, C, reuse_a, reuse_b)
      acc = __builtin_amdgcn_wmma_f32_16x16x4_f32(false, A, false, Bv[c],
                                                  (short)0, acc, false, false);
      // C layout: lane holds N=col; VGPR i holds M=i (lanes 0-15) / M=i+8 (lanes 16-31)
#pragma unroll
      for (int i = 0; i < 8; ++i) pf[i] += fmaxf(acc[i], 0.0f) * w2n[c];
    }

    // reduce over N (16 lanes within each half-wave)
#pragma unroll
    for (int i = 0; i < 8; ++i) {
      float v = pf[i];
      v += __shfl_xor(v, 8, 16);
      v += __shfl_xor(v, 4, 16);
      v += __shfl_xor(v, 2, 16);
      v += __shfl_xor(v, 1, 16);
      pf[i] = v;
    }

    if (col == 0) {                      // lanes 0 (M=0..7) and 16 (M=8..15)
      const int mbase = e0 + (lo ? 0 : 8);
#pragma unroll
      for (int i = 0; i < 8; ++i) {
        int e = mbase + i;
        float pi = pf[i] + b2s;
        unsigned long long key =
            ((unsigned long long)flip_desc(pi) << 32) | (unsigned)(e & (E_PER - 1));
        keys[e] = key;
      }
    }
  }
}

// ---------------- bitonic sort (per 65536-key row, ascending by u64 key) ----------------
// Phase 1: all levels k=2..2048 fused in LDS (partners stay inside a 2048-aligned tile
// since j <= 1024; direction depends on row-local index, carried in from global).
__global__ void bitonic_local_sort(unsigned long long* __restrict__ keys) {
  __shared__ unsigned long long sh[SORT_TILE];
  const int tid = threadIdx.x;
  const size_t gbase = (size_t)blockIdx.x * SORT_TILE;       // tiles never straddle rows
  const int row_off = (int)(gbase & (E_PER - 1));            // offset within row
#pragma unroll
  for (int q = 0; q < SORT_TILE / 256; ++q)
    sh[tid + 256 * q] = keys[gbase + tid + 256 * q];
  __syncthreads();
  for (int k = 2; k <= SORT_TILE; k <<= 1) {
    for (int j = k >> 1; j > 0; j >>= 1) {
      for (int q = 0; q < SORT_TILE / 256; ++q) {
        int t = tid + 256 * q;
        int ixj = t ^ j;
        if (ixj > t) {
          unsigned long long a = sh[t], b = sh[ixj];
          bool asc = ((row_off + t) & k) == 0;               // disjoint bits: exact
          if (asc ? (a > b) : (a < b)) { sh[t] = b; sh[ixj] = a; }
        }
      }
      __syncthreads();
    }
  }
#pragma unroll
  for (int q = 0; q < SORT_TILE / 256; ++q)
    keys[gbase + tid + 256 * q] = sh[tid + 256 * q];
}

// Phase 2a: global compare-exchange pass (only for j >= SORT_TILE).
__global__ void bitonic_global_pass(unsigned long long* __restrict__ keys, int k, int j) {
  int g = blockIdx.x * blockDim.x + threadIdx.x;     // 0..E_TOT-1
  int t = g & (E_PER - 1);
  int ixj = t ^ j;
  if (ixj <= t) return;
  unsigned long long* base = keys + ((size_t)(g >> LOG_EPER) << LOG_EPER);
  unsigned long long a = base[t];
  unsigned long long b = base[ixj];
  bool asc = (t & k) == 0;
  if (asc ? (a > b) : (a < b)) { base[t] = b; base[ixj] = a; }
}

// Phase 2b: for a given k > SORT_TILE, all j = SORT_TILE/2 .. 1 fused in LDS.
// Direction (t & k) is uniform across the tile -> hoisted.
__global__ void bitonic_local_merge(unsigned long long* __restrict__ keys, int k) {
  __shared__ unsigned long long sh[SORT_TILE];
  const int tid = threadIdx.x;
  const size_t gbase = (size_t)blockIdx.x * SORT_TILE;
  const int row_off = (int)(gbase & (E_PER - 1));
  const bool asc = (row_off & k) == 0;
#pragma unroll
  for (int q = 0; q < SORT_TILE / 256; ++q)
    sh[tid + 256 * q] = keys[gbase + tid + 256 * q];
  __syncthreads();
  for (int j = SORT_TILE >> 1; j > 0; j >>= 1) {
    for (int q = 0; q < SORT_TILE / 256; ++q) {
      int t = tid + 256 * q;
      int ixj = t ^ j;
      if (ixj > t) {
        unsigned long long a = sh[t], b = sh[ixj];
        if (asc ? (a > b) : (a < b)) { sh[t] = b; sh[ixj] = a; }
      }
    }
    __syncthreads();
  }
#pragma unroll
  for (int q = 0; q < SORT_TILE / 256; ++q)
    keys[gbase + tid + 256 * q] = sh[tid + 256 * q];
}

// ---------------- gather top-K edges (rank order == descending score) ----------------
__global__ void gather_topk(const unsigned long long* __restrict__ keys,
                            const int* __restrict__ edge_index,
                            const float* __restrict__ edge_attr,
                            int* __restrict__ ei_raw,
                            float* __restrict__ ea_out) {
  int s = blockIdx.x * blockDim.x + threadIdx.x;     // 0..BK-1
  if (s >= BK) return;
  int row  = s >> LOG_K;
  int rank = s & (KK - 1);
  unsigned long long key = keys[((size_t)row << LOG_EPER) + rank];
  int idx = (int)(unsigned)(key & 0xFFFFFFFFull);    // index within row
  int g = (row << LOG_EPER) + idx;                   // global edge id
  ei_raw[s]      = edge_index[g];                    // edge_index[0, g]
  ei_raw[BK + s] = edge_index[E_TOT + g];            // edge_index[1, g]
  ea_out[2 * s + 0] = edge_attr[2 * g + 0];
  ea_out[2 * s + 1] = edge_attr[2 * g + 1];
}

__global__ void mark_nodes(const int* __restrict__ ei_raw, int* __restrict__ mask, int n) {
  int i = blockIdx.x * blockDim.x + threadIdx.x;
  if (i < n) mask[ei_raw[i]] = 1;                    // benign race, all write 1
}

// ---------------- inclusive scan of mask -> newid (3 kernels, 1024 elems/block) ----------------
__global__ void scan_pass1(const int* __restrict__ mask, int* __restrict__ newid,
                           int* __restrict__ bsums) {
  __shared__ int sh[256];
  int tid = threadIdx.x, bid = blockIdx.x;
  int base = bid * 1024 + tid * 4;
  int a0 = mask[base], a1 = mask[base + 1], a2 = mask[base + 2], a3 = mask[base + 3];
  int s1 = a0 + a1, s2 = s1 + a2, s3 = s2 + a3;
  sh[tid] = s3;
  __syncthreads();
  for (int off = 1; off < 256; off <<= 1) {
    int v = (tid >= off) ? sh[tid - off] : 0;
    __syncthreads();
    sh[tid] += v;
    __syncthreads();
  }
  int excl = sh[tid] - s3;
  newid[base]     = excl + a0;
  newid[base + 1] = excl + s1;
  newid[base + 2] = excl + s2;
  newid[base + 3] = excl + s3;
  if (tid == 255) bsums[bid] = sh[255];
}

__global__ void scan_pass2(int* __restrict__ bsums) {   // 256 entries -> exclusive scan
  __shared__ int sh[256];
  int tid = threadIdx.x;
  int orig = bsums[tid];
  sh[tid] = orig;
  __syncthreads();
  for (int off = 1; off < 256; off <<= 1) {
    int v = (tid >= off) ? sh[tid - off] : 0;
    __syncthreads();
    sh[tid] += v;
    __syncthreads();
  }
  bsums[tid] = sh[tid] - orig;
}

__global__ void scan_pass3(int* __restrict__ newid, const int* __restrict__ bsums) {
  int i = blockIdx.x * blockDim.x + threadIdx.x;     // NN threads
  newid[i] = newid[i] + bsums[i >> 10] - 1;          // cumsum - 1
}

__global__ void relabel(const int* __restrict__ ei_raw, const int* __restrict__ newid,
                        int* __restrict__ ei_out, int n) {
  int i = blockIdx.x * blockDim.x + threadIdx.x;
  if (i < n) ei_out[i] = newid[ei_raw[i]];
}

// ---------------- host launch ----------------
extern "C" void kernel_launch(void* const* d_in, const int* in_sizes, int n_in,
                              void* d_out, int out_size, void* d_ws, size_t ws_size,
                              hipStream_t stream) {
  const float* x          = (const float*)d_in[0];
  const int*   edge_index = (const int*)  d_in[1];
  const float* edge_attr  = (const float*)d_in[2];
  const int*   batch      = (const int*)  d_in[3];
  // d_in[4] = edge_batch (unused: implied by layout)
  const float* w1 = (const float*)d_in[5];
  const float* b1 = (const float*)d_in[6];
  const float* w2 = (const float*)d_in[7];
  const float* b2 = (const float*)d_in[8];

  // output layout (flat, element = 4 bytes): x | ei(2,BK) | ea(BK,2) | batch
  float* out_x     = (float*)d_out;
  int*   out_ei    = (int*)((float*)d_out + (size_t)NN * DD);
  float* out_ea    = (float*)d_out + (size_t)NN * DD + 2 * (size_t)BK;
  int*   out_batch = (int*)((float*)d_out + (size_t)NN * DD + 4 * (size_t)BK);

  // workspace layout
  uint8_t* ws = (uint8_t*)d_ws;
  unsigned long long* keys = (unsigned long long*)ws;                 // E_TOT * 8  = 8 MiB
  int* ei_raw = (int*)(ws + (size_t)E_TOT * 8);                       // 2*BK * 4   = 4 MiB
  int* mask   = (int*)(ws + (size_t)E_TOT * 8 + (size_t)2 * BK * 4);  // NN * 4     = 1 MiB
  int* newid  = (int*)((uint8_t*)mask + (size_t)NN * 4);              // NN * 4     = 1 MiB
  int* bsums  = (int*)((uint8_t*)newid + (size_t)NN * 4);             // 256 * 4

  // 1) pass-through copies (x dominates: 128 MB each way ~ 11 us at 23.3 TB/s)
  copy_f4<<<(NN * DD / 4 + 255) / 256, 256, 0, stream>>>((const float4*)x, (float4*)out_x, NN * DD / 4);
  copy_i32<<<(NN + 255) / 256, 256, 0, stream>>>(batch, out_batch, NN);

  // 2) edge MLP -> sortable keys (WMMA f32 16x16x4)
  {
    int waves = E_TOT / (16 * TPW);                  // 8192 waves
    int blocks = waves * 32 / 256;                   // 1024 blocks of 8 waves
    edge_mlp_wmma<<<blocks, 256, 0, stream>>>(edge_attr, w1, b1, w2, b2, keys);
  }

  // 3) bitonic sort each 65536-key row ascending (== descending score, asc index ties)
  //    LDS-fused: 21 launches instead of 136, ~6x less global key traffic.
  bitonic_local_sort<<<E_TOT / SORT_TILE, 256, 0, stream>>>(keys);
  for (int k = SORT_TILE * 2; k <= E_PER; k <<= 1) {
    for (int j = k >> 1; j >= SORT_TILE; j >>= 1)
      bitonic_global_pass<<<E_TOT / 256, 256, 0, stream>>>(keys, k, j);
    bitonic_local_merge<<<E_TOT / SORT_TILE, 256, 0, stream>>>(keys, k);
  }

  // 4) gather selected edges
  gather_topk<<<(BK + 255) / 256, 256, 0, stream>>>(keys, edge_index, edge_attr, ei_raw, out_ea);

  // 5) node mask + inclusive scan -> newid
  zero_i32<<<(NN + 255) / 256, 256, 0, stream>>>(mask, NN);
  mark_nodes<<<(2 * BK + 255) / 256, 256, 0, stream>>>(ei_raw, mask, 2 * BK);
  scan_pass1<<<NN / 1024, 256, 0, stream>>>(mask, newid, bsums);
  scan_pass2<<<1, 256, 0, stream>>>(bsums);
  scan_pass3<<<NN / 256, 256, 0, stream>>>(newid, bsums);

  // 6) relabel edge indices
  relabel<<<(2 * BK + 255) / 256, 256, 0, stream>>>(ei_raw, newid, out_ei, 2 * BK);
}